// Classifier_34660386078864
// MI455X (gfx1250) — compile-verified
//
#include <hip/hip_runtime.h>
#include <hip/hip_bf16.h>

#define Nn 80000
#define Ee 640000
#define Bg 800
#define NPGc 100
#define HIDc 128
#define NCc 10
#define INDc 20
#define EPSf 1e-5f

typedef __attribute__((ext_vector_type(2))) float v2f;
typedef __attribute__((ext_vector_type(8))) float v8f;

// ---------------- utility ----------------
__global__ __launch_bounds__(256) void k_zero(float* p, long long n) {
  long long i = (long long)blockIdx.x * blockDim.x + threadIdx.x;
  long long st = (long long)gridDim.x * blockDim.x;
  for (; i < n; i += st) p[i] = 0.f;
}

// ---------------- degrees ----------------
__global__ __launch_bounds__(256) void k_deg(const int* __restrict__ src, const int* __restrict__ dst,
                                             float* deg_in, float* deg_out, int E) {
  int e = blockIdx.x * blockDim.x + threadIdx.x;
  if (e >= E) return;
  atomicAdd(&deg_in[dst[e]], 1.0f);
  atomicAdd(&deg_out[src[e]], 1.0f);
}

__global__ __launch_bounds__(256) void k_inv(const float* deg_in, const float* deg_out,
                                             float* inv_in, float* inv_out, int n) {
  int i = blockIdx.x * blockDim.x + threadIdx.x;
  if (i >= n) return;
  inv_in[i]  = rsqrtf(fmaxf(deg_in[i], 1.0f));
  inv_out[i] = rsqrtf(fmaxf(deg_out[i], 1.0f));
}

// gconv1: one-hot input -> single atomic per edge into N x 20 histogram
__global__ __launch_bounds__(256) void k_edge_onehot(const int* __restrict__ src, const int* __restrict__ dst,
                                                     const float* __restrict__ deg_in,
                                                     const float* __restrict__ inv_out,
                                                     float* agg, int E) {
  int e = blockIdx.x * blockDim.x + threadIdx.x;
  if (e >= E) return;
  int s = src[e];
  int c = (int)deg_in[s];
  c = c > (INDc - 1) ? (INDc - 1) : c;
  atomicAdd(&agg[(size_t)dst[e] * INDc + c], inv_out[s]);
}

// gconv2 message pass: gather h1[src]*inv_out[src], scatter-add to agg[dst]
__global__ __launch_bounds__(256) void k_edge_gs(const int* __restrict__ src, const int* __restrict__ dst,
                                                 const float* __restrict__ inv_out,
                                                 const float* __restrict__ H, float* agg, int E) {
  long long idx = (long long)blockIdx.x * 256 + threadIdx.x;
  if (idx >= (long long)E * 32) return;
  int e  = (int)(idx >> 5);
  int f0 = ((int)idx & 31) * 4;
  int s = src[e], d = dst[e];
  float w = inv_out[s];
  float4 v = *(const float4*)(H + (size_t)s * HIDc + f0);
  float* a = agg + (size_t)d * HIDc + f0;
  atomicAdd(a + 0, v.x * w);
  atomicAdd(a + 1, v.y * w);
  atomicAdd(a + 2, v.z * w);
  atomicAdd(a + 3, v.w * w);
}

// ---------------- fp32 WMMA GEMM: out = (A * rowscale) @ W + bias ----------------
// A: M x K (row-major), W: K x 128 (row-major), out: M x 128. K multiple of 4.
// Dynamic LDS: alds[128*K] (A tile, rowscaled) + blds[K*128] (W in fragment-
// native pair layout: blds[(k/2)*256 + 2n + (k&1)] = W[k][n]) so every A/B
// fragment is one aligned ds_load_b64.
__global__ __launch_bounds__(256) void k_gemm_wmma(const float* __restrict__ A,
                                                   const float* __restrict__ W,
                                                   const float* __restrict__ bias,
                                                   const float* __restrict__ rowscale,
                                                   float* __restrict__ out, int M, int K) {
  extern __shared__ float smem[];
  float* alds = smem;            // 128 * K
  float* blds = smem + 128 * K;  // K * 128
  int tid = threadIdx.x;
  int rblk = blockIdx.x * 128;

  // stage A tile (coalesced, rowscale fused, zero-padded past M)
  for (int i = tid; i < 128 * K; i += 256) {
    int r = i / K;
    int gr = rblk + r;
    float v = 0.f;
    if (gr < M) {
      float rs = rowscale ? rowscale[gr] : 1.0f;
      v = A[(size_t)gr * K + (i - r * K)] * rs;
    }
    alds[i] = v;
  }
  // stage W in pair-interleaved layout
  for (int i = tid; i < K * HIDc; i += 256) {
    int k = i >> 7;          // i / 128
    int n = i & (HIDc - 1);  // i % 128
    blds[(k >> 1) * 256 + 2 * n + (k & 1)] = W[i];
  }
  __syncthreads();

  int wave = tid >> 5, lane = tid & 31;
  int m = lane & 15, khalf = lane >> 4;
  const float* arow = alds + (size_t)(wave * 16 + m) * K;

  v8f acc[8] = {};
  for (int k0 = 0; k0 < K; k0 += 4) {
    v2f a = *(const v2f*)(arow + k0 + 2 * khalf);
    const float* bp = blds + (k0 / 2 + khalf) * 256 + 2 * m;
#pragma unroll
    for (int nt = 0; nt < 8; ++nt) {
      v2f b = *(const v2f*)(bp + nt * 32);
      acc[nt] = __builtin_amdgcn_wmma_f32_16x16x4_f32(false, a, false, b,
                                                      (short)0, acc[nt], false, false);
    }
  }

  int orow_base = rblk + wave * 16 + khalf * 8;
#pragma unroll
  for (int nt = 0; nt < 8; ++nt) {
    int col = nt * 16 + m;
    float bv = bias ? bias[col] : 0.f;
#pragma unroll
    for (int r = 0; r < 8; ++r) {
      int orow = orow_base + r;
      if (orow < M) out[(size_t)orow * HIDc + col] = acc[nt][r] + bv;
    }
  }
}

// ---------------- batch norm ----------------
__global__ __launch_bounds__(128) void k_bnstat(const float* __restrict__ X, int M, float* stats) {
  int c = threadIdx.x;
  int r0 = blockIdx.x * 256;
  int rend = r0 + 256; if (rend > M) rend = M;
  float s = 0.f, s2 = 0.f;
  for (int r = r0; r < rend; ++r) {
    float v = X[(size_t)r * HIDc + c];
    s += v; s2 += v * v;
  }
  atomicAdd(&stats[c], s);
  atomicAdd(&stats[HIDc + c], s2);
}

__global__ __launch_bounds__(128) void k_bnfin(const float* __restrict__ gamma,
                                               const float* __restrict__ beta,
                                               float* stats, int M) {
  int c = threadIdx.x;
  float mean = stats[c] / (float)M;
  float var  = stats[HIDc + c] / (float)M - mean * mean;
  float a = gamma[c] * rsqrtf(var + EPSf);
  stats[2 * HIDc + c] = a;
  stats[3 * HIDc + c] = beta[c] - mean * a;
}

// mode 0: ELU, mode 1: tanh
__global__ __launch_bounds__(256) void k_bnapply(float* X, const float* __restrict__ stats,
                                                 long long n, int mode) {
  long long i = (long long)blockIdx.x * 256 + threadIdx.x;
  if (i >= n) return;
  int c = (int)(i & (HIDc - 1));
  float y = X[i] * stats[2 * HIDc + c] + stats[3 * HIDc + c];
  y = (mode == 0) ? (y > 0.f ? y : expf(y) - 1.f) : tanhf(y);
  X[i] = y;
}

// ---------------- value head + pooling ----------------
__global__ __launch_bounds__(256) void k_dotWv(const float* __restrict__ H2,
                                               const float* __restrict__ Wv, float* p, int n) {
  int i = blockIdx.x * blockDim.x + threadIdx.x;
  if (i >= n) return;
  const float4* row = (const float4*)(H2 + (size_t)i * HIDc);
  const float4* w4  = (const float4*)Wv;
  float s = 0.f;
#pragma unroll 8
  for (int k = 0; k < HIDc / 4; ++k) {
    float4 a = row[k], b = w4[k];
    s += a.x * b.x + a.y * b.y + a.z * b.z + a.w * b.w;
  }
  p[i] = s;
}

__global__ __launch_bounds__(256) void k_edgev(const int* __restrict__ src, const int* __restrict__ dst,
                                               const float* __restrict__ inv_out,
                                               const float* __restrict__ p, float* dacc, int E) {
  int e = blockIdx.x * blockDim.x + threadIdx.x;
  if (e >= E) return;
  int s = src[e];
  atomicAdd(&dacc[dst[e]], inv_out[s] * p[s]);
}

__global__ __launch_bounds__(256) void k_finald(float* dacc, const float* __restrict__ inv_in,
                                                const float* __restrict__ bvp, int n) {
  int i = blockIdx.x * blockDim.x + threadIdx.x;
  if (i >= n) return;
  dacc[i] = dacc[i] * inv_in[i] + bvp[0];
}

__global__ __launch_bounds__(128) void k_pool(const float* __restrict__ H2,
                                              const float* __restrict__ d, float* vec) {
  int g = blockIdx.x, c = threadIdx.x;
  int base = g * NPGc;
  float s = 0.f;
  for (int i = 0; i < NPGc; ++i) s += d[base + i] * H2[(size_t)(base + i) * HIDc + c];
  vec[(size_t)g * HIDc + c] = s;
}

__global__ __launch_bounds__(256) void k_sumall(const float* __restrict__ x, int n, float* acc) {
  __shared__ float sh[256];
  int tid = threadIdx.x;
  float s = 0.f;
  for (int i = blockIdx.x * 256 + tid; i < n; i += gridDim.x * 256) s += x[i];
  sh[tid] = s;
  __syncthreads();
  for (int o = 128; o > 0; o >>= 1) { if (tid < o) sh[tid] += sh[tid + o]; __syncthreads(); }
  if (tid == 0) atomicAdd(acc, sh[0]);
}

__global__ __launch_bounds__(256) void k_shrink(const float* __restrict__ vec,
                                                const float* __restrict__ acc,
                                                float* hg, int n) {
  int i = blockIdx.x * blockDim.x + threadIdx.x;
  if (i >= n) return;
  float mean = acc[0] / (float)n;
  float x = vec[i];
  float y = fmaxf(x - mean, 0.f) - fmaxf(-x - mean, 0.f);
  hg[i] = tanhf(y);
}

// ---------------- classifier head ----------------
__global__ __launch_bounds__(32) void k_cls(const float* __restrict__ hg,
                                            const float* __restrict__ Wc,
                                            const float* __restrict__ bc, float* out) {
  int row = blockIdx.x;
  int col = threadIdx.x;
  if (col >= NCc) return;
  float s = bc[col];
  const float* r = hg + (size_t)row * HIDc;
  for (int k = 0; k < HIDc; ++k) s += r[k] * Wc[k * NCc + col];
  out[row * NCc + col] = tanhf(s);
}

// ---------------- host launcher ----------------
extern "C" void kernel_launch(void* const* d_in, const int* in_sizes, int n_in,
                              void* d_out, int out_size, void* d_ws, size_t ws_size,
                              hipStream_t stream) {
  const int* src = (const int*)d_in[0];
  const int* dst = (const int*)d_in[1];
  const float* W1  = (const float*)d_in[3];
  const float* b1  = (const float*)d_in[4];
  const float* g1  = (const float*)d_in[5];
  const float* be1 = (const float*)d_in[6];
  const float* W2  = (const float*)d_in[7];
  const float* b2  = (const float*)d_in[8];
  const float* g2  = (const float*)d_in[9];
  const float* be2 = (const float*)d_in[10];
  const float* Wv  = (const float*)d_in[11];
  const float* bvp = (const float*)d_in[12];
  const float* Wg  = (const float*)d_in[15];
  const float* bgv = (const float*)d_in[16];
  const float* g3  = (const float*)d_in[17];
  const float* be3 = (const float*)d_in[18];
  const float* Wc  = (const float*)d_in[19];
  const float* bc  = (const float*)d_in[20];
  float* out = (float*)d_out;
  float* ws  = (float*)d_ws;

  const size_t N = Nn, B = Bg;
  size_t off = 0;
  size_t o_deg_in  = off; off += N;
  size_t o_deg_out = off; off += N;
  size_t o_agg     = off; off += N * HIDc;
  size_t o_dacc    = off; off += N;
  size_t o_stats   = off; off += 3 * 4 * HIDc;   // 3 BN stages x (sum,sumsq,a,b)
  size_t o_mean    = off; off += 1;
  off = (off + 3) & ~(size_t)3;
  const long long zcount = (long long)off;       // front region zeroed each call
  size_t o_inv_in  = off; off += N;
  size_t o_inv_out = off; off += N;
  size_t o_H1      = off; off += N * HIDc;
  size_t o_H2      = off; off += N * HIDc;
  size_t o_p       = off; off += N;
  size_t o_vec     = off; off += B * HIDc;
  size_t o_hg      = off; off += B * HIDc;
  (void)ws_size; (void)n_in; (void)in_sizes; (void)out_size;

  float* st1 = ws + o_stats;
  float* st2 = ws + o_stats + 4 * HIDc;
  float* st3 = ws + o_stats + 8 * HIDc;

  const size_t smem20  = (size_t)(2 * 128 * INDc) * sizeof(float);   // 20 KB
  const size_t smem128 = (size_t)(2 * 128 * HIDc) * sizeof(float);   // 128 KB

  // 0. zero accumulators
  k_zero<<<(int)((zcount + 255) / 256), 256, 0, stream>>>(ws, zcount);

  // 1. degrees, inverse sqrt degrees
  k_deg<<<(Ee + 255) / 256, 256, 0, stream>>>(src, dst, ws + o_deg_in, ws + o_deg_out, Ee);
  k_inv<<<(Nn + 255) / 256, 256, 0, stream>>>(ws + o_deg_in, ws + o_deg_out,
                                              ws + o_inv_in, ws + o_inv_out, Nn);

  // 2. gconv1 (one-hot collapse) -> H1, BN+ELU
  k_edge_onehot<<<(Ee + 255) / 256, 256, 0, stream>>>(src, dst, ws + o_deg_in,
                                                      ws + o_inv_out, ws + o_agg, Ee);
  k_gemm_wmma<<<(Nn + 127) / 128, 256, smem20, stream>>>(ws + o_agg, W1, b1, ws + o_inv_in,
                                                         ws + o_H1, Nn, INDc);
  k_bnstat<<<(Nn + 255) / 256, 128, 0, stream>>>(ws + o_H1, Nn, st1);
  k_bnfin<<<1, 128, 0, stream>>>(g1, be1, st1, Nn);
  k_bnapply<<<(int)(((long long)Nn * HIDc + 255) / 256), 256, 0, stream>>>(
      ws + o_H1, st1, (long long)Nn * HIDc, 0);

  // 3. gconv2 -> H2, BN+ELU
  k_zero<<<(int)(((long long)Nn * HIDc + 255) / 256), 256, 0, stream>>>(ws + o_agg,
                                                                        (long long)Nn * HIDc);
  k_edge_gs<<<(int)(((long long)Ee * 32 + 255) / 256), 256, 0, stream>>>(
      src, dst, ws + o_inv_out, ws + o_H1, ws + o_agg, Ee);
  k_gemm_wmma<<<(Nn + 127) / 128, 256, smem128, stream>>>(ws + o_agg, W2, b2, ws + o_inv_in,
                                                          ws + o_H2, Nn, HIDc);
  k_bnstat<<<(Nn + 255) / 256, 128, 0, stream>>>(ws + o_H2, Nn, st2);
  k_bnfin<<<1, 128, 0, stream>>>(g2, be2, st2, Nn);
  k_bnapply<<<(int)(((long long)Nn * HIDc + 255) / 256), 256, 0, stream>>>(
      ws + o_H2, st2, (long long)Nn * HIDc, 0);

  // 4. value head: p = H2 @ Wv folded through the message pass (1 atomic/edge)
  k_dotWv<<<(Nn + 255) / 256, 256, 0, stream>>>(ws + o_H2, Wv, ws + o_p, Nn);
  k_edgev<<<(Ee + 255) / 256, 256, 0, stream>>>(src, dst, ws + o_inv_out, ws + o_p,
                                                ws + o_dacc, Ee);
  k_finald<<<(Nn + 255) / 256, 256, 0, stream>>>(ws + o_dacc, ws + o_inv_in, bvp, Nn);

  // 5. weighted graph pooling (contiguous graph blocks), soft-shrink, tanh
  k_pool<<<Bg, 128, 0, stream>>>(ws + o_H2, ws + o_dacc, ws + o_vec);
  k_sumall<<<(Bg * HIDc + 255) / 256, 256, 0, stream>>>(ws + o_vec, Bg * HIDc, ws + o_mean);
  k_shrink<<<(Bg * HIDc + 255) / 256, 256, 0, stream>>>(ws + o_vec, ws + o_mean,
                                                        ws + o_hg, Bg * HIDc);

  // 6. hg @ Wg + bg -> BN -> tanh (result kept in o_vec)
  k_gemm_wmma<<<(Bg + 127) / 128, 256, smem128, stream>>>(ws + o_hg, Wg, bgv, nullptr,
                                                          ws + o_vec, Bg, HIDc);
  k_bnstat<<<(Bg + 255) / 256, 128, 0, stream>>>(ws + o_vec, Bg, st3);
  k_bnfin<<<1, 128, 0, stream>>>(g3, be3, st3, Bg);
  k_bnapply<<<(Bg * HIDc + 255) / 256, 256, 0, stream>>>(ws + o_vec, st3,
                                                         (long long)Bg * HIDc, 1);

  // 7. classifier: tanh(hg @ Wc + bc)
  k_cls<<<Bg, 32, 0, stream>>>(ws + o_vec, Wc, bc, out);
}